// CascadeXML_32865089749354
// MI455X (gfx1250) — compile-verified
//
#include <hip/hip_runtime.h>
#include <hip/hip_bf16.h>
#include <math.h>

// ---- problem constants (match reference) ----
#define BATCH 64
#define EMB   768
#define N0    2048
#define N1    16384
#define NL    131072
#define CS    8
#define KTOP  50
#define NC    (KTOP * CS)        // 400 candidates per level
#define OUTW  (N0 + NC + NC)     // 2848 output cols per row

typedef float v2f __attribute__((ext_vector_type(2)));
typedef float v8f __attribute__((ext_vector_type(8)));

__device__ __forceinline__ float sigmoidf_(float x) {
    return 1.0f / (1.0f + expf(-x));
}

// ---------------------------------------------------------------------------
// Dense GEMM via V_WMMA_F32_16X16X4_F32 (fp32 matrix pipe, full precision).
//   out[m,n] = sum_k A[m,k] * W[n,k] + bias[n]   (W row-major [N,K], i.e. @W.T)
// One wave (32 threads) computes one 16x16 tile, looping K in steps of 4.
// A 16x4 layout (ISA 7.12.2): lane l -> M = l&15, holds K = (l>>4)*2 + {0,1}.
// B 4x16 mirrors with N = l&15. C/D: row = v + (l>=16 ? 8:0), col = l&15.
// ---------------------------------------------------------------------------
template <bool SIGMOID>
__global__ void gemm_wmma_f32(const float* __restrict__ A, int lda,
                              const float* __restrict__ W, int ldw,
                              const float* __restrict__ bias,
                              float* __restrict__ out, int ldo, int K) {
    const int lane  = threadIdx.x & 31;
    const int tileN = blockIdx.x * 16;
    const int tileM = blockIdx.y * 16;
    const int mn    = lane & 15;          // M index for A, N index for B
    const int kb    = (lane >> 4) * 2;    // K sub-offset for this half-wave

    const float* aptr = A + (size_t)(tileM + mn) * lda + kb;
    const float* bptr = W + (size_t)(tileN + mn) * ldw + kb;

    v8f c = {0.f, 0.f, 0.f, 0.f, 0.f, 0.f, 0.f, 0.f};
#pragma unroll 4
    for (int k = 0; k < K; k += 4) {
        v2f a, b;
        a.x = aptr[k];
        a.y = aptr[k + 1];
        b.x = bptr[k];
        b.y = bptr[k + 1];
        // D = A(16x4) x B(4x16) + C, fp32 in / fp32 accum
        c = __builtin_amdgcn_wmma_f32_16x16x4_f32(
                /*neg_a=*/false, a, /*neg_b=*/false, b,
                /*c_mod=*/(short)0, c, /*reuse_a=*/false, /*reuse_b=*/false);
    }

    const int col   = tileN + mn;
    const int rbase = tileM + ((lane >> 4) << 3);
    const float bv  = bias[col];
#pragma unroll
    for (int v = 0; v < 8; ++v) {
        float x = c[v] + bv;
        if (SIGMOID) x = sigmoidf_(x);
        out[(size_t)(rbase + v) * ldo + col] = x;
    }
}

// ---------------------------------------------------------------------------
// Iterative top-k per row (selection of max, k times). One block per row.
// Tie-break: lower index wins (matches jax.lax.top_k). Values >= 0 (sigmoid),
// so -1.0f works as the "consumed" sentinel.
// ---------------------------------------------------------------------------
__global__ void topk_rows(const float* __restrict__ src, int srcStride, int n,
                          float* __restrict__ vals, int* __restrict__ idxs,
                          int k) {
    __shared__ float s_v[2048];
    __shared__ float r_v[256];
    __shared__ int   r_i[256];

    const int row = blockIdx.x;
    const int tid = threadIdx.x;
    const float* p = src + (size_t)row * srcStride;

    for (int i = tid; i < n; i += blockDim.x) s_v[i] = p[i];
    __syncthreads();

    for (int t = 0; t < k; ++t) {
        float bv = -1.0f;
        int   bi = 0x7fffffff;
        for (int i = tid; i < n; i += blockDim.x) {
            float v = s_v[i];
            if (v > bv || (v == bv && i < bi)) { bv = v; bi = i; }
        }
        r_v[tid] = bv; r_i[tid] = bi;
        __syncthreads();
        for (int off = 128; off > 0; off >>= 1) {
            if (tid < off) {
                float ov = r_v[tid + off];
                int   oi = r_i[tid + off];
                if (ov > r_v[tid] || (ov == r_v[tid] && oi < r_i[tid])) {
                    r_v[tid] = ov; r_i[tid] = oi;
                }
            }
            __syncthreads();
        }
        if (tid == 0) {
            vals[row * k + t] = r_v[0];
            idxs[row * k + t] = r_i[0];
            s_v[r_i[0]] = -1.0f;   // mask out
        }
        __syncthreads();
    }
}

// ---------------------------------------------------------------------------
// Level-1 scoring: one wave per (batch, candidate). 400 candidates/row.
//   cand = clusters0[idx1[b,j]*8 + s]; logit = C1[cand]·feat1[b] + b1[cand]
//   w1 = sigmoid(logit) * vals1[b,j]
// float4 gather loads (global_load_b128), wave32 xor-shuffle reduction.
// ---------------------------------------------------------------------------
__global__ void score_level1(const float* __restrict__ C1,
                             const float* __restrict__ b1,
                             const float* __restrict__ feat1,
                             const int*   __restrict__ clusters0,
                             const int*   __restrict__ idx1,
                             const float* __restrict__ vals1,
                             float* __restrict__ out,
                             float* __restrict__ probs1,
                             int*   __restrict__ cand1) {
    const int gw   = (blockIdx.x * blockDim.x + threadIdx.x) >> 5;
    const int lane = threadIdx.x & 31;
    const int b = gw / NC;
    const int c = gw % NC;
    const int j = c >> 3, s = c & 7;

    const int   parent  = idx1[b * KTOP + j];
    const float pval    = vals1[b * KTOP + j];
    const int   cand    = clusters0[parent * CS + s];

    const float4* crow = (const float4*)(C1 + (size_t)cand * EMB);
    const float4* frow = (const float4*)(feat1 + (size_t)b * EMB);
    float sum = 0.f;
#pragma unroll
    for (int i = 0; i < EMB / 128; ++i) {   // 6 iters: 192 float4 / 32 lanes
        float4 cv = crow[lane + i * 32];
        float4 fv = frow[lane + i * 32];
        sum += cv.x * fv.x + cv.y * fv.y + cv.z * fv.z + cv.w * fv.w;
    }
#pragma unroll
    for (int off = 16; off > 0; off >>= 1) sum += __shfl_xor(sum, off, 32);

    if (lane == 0) {
        float logit = sum + b1[cand];
        float p = sigmoidf_(logit);
        out[(size_t)b * OUTW + N0 + c] = p * pval;
        probs1[b * NC + c] = p;
        cand1[b * NC + c]  = cand;
    }
}

// ---------------------------------------------------------------------------
// Level-2 scoring: idx2 selects among the 400 level-1 candidates.
//   ind2 = cand1[b, idx2[b,j]]; cand = clusters1[ind2*8+s]
//   logit = C2[cand]·feat2[b] + b2[cand]; p = (logit==0) ? 0 : sigmoid(logit)
// ---------------------------------------------------------------------------
__global__ void score_level2(const float* __restrict__ C2,
                             const float* __restrict__ b2,
                             const float* __restrict__ feat2,
                             const int*   __restrict__ clusters1,
                             const int*   __restrict__ cand1,
                             const int*   __restrict__ idx2,
                             const float* __restrict__ vals2,
                             float* __restrict__ out) {
    const int gw   = (blockIdx.x * blockDim.x + threadIdx.x) >> 5;
    const int lane = threadIdx.x & 31;
    const int b = gw / NC;
    const int c = gw % NC;
    const int j = c >> 3, s = c & 7;

    const int   sel  = idx2[b * KTOP + j];       // index into 400 candidates
    const float pval = vals2[b * KTOP + j];
    const int   ind2 = cand1[b * NC + sel];      // level-1 id
    const int   cand = clusters1[ind2 * CS + s]; // label id

    const float4* crow = (const float4*)(C2 + (size_t)cand * EMB);
    const float4* frow = (const float4*)(feat2 + (size_t)b * EMB);
    float sum = 0.f;
#pragma unroll
    for (int i = 0; i < EMB / 128; ++i) {
        float4 cv = crow[lane + i * 32];
        float4 fv = frow[lane + i * 32];
        sum += cv.x * fv.x + cv.y * fv.y + cv.z * fv.z + cv.w * fv.w;
    }
#pragma unroll
    for (int off = 16; off > 0; off >>= 1) sum += __shfl_xor(sum, off, 32);

    if (lane == 0) {
        float logit = sum + b2[cand];
        float p = (logit == 0.0f) ? 0.0f : sigmoidf_(logit);
        out[(size_t)b * OUTW + N0 + NC + c] = p * pval;
    }
}

// ---------------------------------------------------------------------------
extern "C" void kernel_launch(void* const* d_in, const int* in_sizes, int n_in,
                              void* d_out, int out_size, void* d_ws, size_t ws_size,
                              hipStream_t stream) {
    const float* feat0     = (const float*)d_in[0];   // [64,1536]
    const float* feat1     = (const float*)d_in[1];   // [64,768]
    const float* feat2     = (const float*)d_in[2];   // [64,768]
    const float* Wh        = (const float*)d_in[3];   // [768,1536]
    const float* bh        = (const float*)d_in[4];   // [768]
    const float* C0        = (const float*)d_in[5];   // [2048,768]
    const float* b0        = (const float*)d_in[6];   // [2048]
    const float* C1        = (const float*)d_in[7];   // [16384,768]
    const float* b1        = (const float*)d_in[8];   // [16384]
    const float* C2        = (const float*)d_in[9];   // [131072,768]
    const float* b2        = (const float*)d_in[10];  // [131072]
    const int*   clusters0 = (const int*)d_in[11];    // [2048,8]
    const int*   clusters1 = (const int*)d_in[12];    // [16384,8]
    float* out = (float*)d_out;                       // [64, 2848]

    // workspace layout (floats/ints, 4B units)
    float* ws     = (float*)d_ws;
    float* h0     = ws;                        // 64*768      = 49152
    float* vals1  = h0 + BATCH * EMB;          // 64*50       =  3200
    int*   idx1   = (int*)(vals1 + BATCH * KTOP);
    int*   cand1  = (int*)((float*)idx1 + BATCH * KTOP); // 64*400
    float* probs1 = (float*)(cand1 + BATCH * NC);        // 64*400
    float* vals2  = probs1 + BATCH * NC;
    int*   idx2   = (int*)(vals2 + BATCH * KTOP);
    // total: 113152 * 4B ~= 452 KB of d_ws

    // 1) h0 = feat0 @ Wh.T + bh            (fp32 WMMA)
    gemm_wmma_f32<false><<<dim3(EMB / 16, BATCH / 16), 32, 0, stream>>>(
        feat0, 2 * EMB, Wh, 2 * EMB, bh, h0, EMB, 2 * EMB);

    // 2) probs0 = sigmoid(h0 @ C0.T + b0) -> out[:, 0:2048]   (fp32 WMMA)
    gemm_wmma_f32<true><<<dim3(N0 / 16, BATCH / 16), 32, 0, stream>>>(
        h0, EMB, C0, EMB, b0, out, OUTW, EMB);

    // 3) top-50 of probs0 (read back from out rows, stride OUTW)
    topk_rows<<<BATCH, 256, 0, stream>>>(out, OUTW, N0, vals1, idx1, KTOP);

    // 4) level-1 gather-dot scoring: 64*400 waves, 8 waves per 256-thr block
    score_level1<<<(BATCH * NC) / 8, 256, 0, stream>>>(
        C1, b1, feat1, clusters0, idx1, vals1, out, probs1, cand1);

    // 5) top-50 of probs1 (n = 400)
    topk_rows<<<BATCH, 256, 0, stream>>>(probs1, NC, NC, vals2, idx2, KTOP);

    // 6) level-2 gather-dot scoring
    score_level2<<<(BATCH * NC) / 8, 256, 0, stream>>>(
        C2, b2, feat2, clusters1, cand1, idx2, vals2, out);

    (void)in_sizes; (void)n_in; (void)out_size; (void)ws_size;
}